// Prediction_54305566491023
// MI455X (gfx1250) — compile-verified
//
#include <hip/hip_runtime.h>

// out[b,i,j] = s[b,i]*e[b,j] on the band i <= j <= i+15, else 0.
// B=32, L=2048, fp32 in/out. Output = 512 MB -> pure store-bandwidth problem.
// Kernel 1: non-temporal B128 zero-fill of the whole output.
// Kernel 2: band tiles via V_WMMA_F32_16X16X4_F32 rank-1 outer products.

typedef __attribute__((ext_vector_type(2))) float v2f;
typedef __attribute__((ext_vector_type(4))) float v4f;
typedef __attribute__((ext_vector_type(8))) float v8f;

#define BATCH 32
#define LEN   2048
#define BAND  15

// ---------------------------------------------------------------------------
// Kernel 1: stream zeros over the full [B, L, L] output.
// 256 threads/block, 4 float4 NT stores per thread -> 512 B per wave-iteration,
// coalesced lanes, grid = total_float4 / 1024 = 32768 blocks.
// ---------------------------------------------------------------------------
__global__ void zero_fill_kernel(v4f* __restrict__ out4) {
    const int tid = threadIdx.x;
    const long long base = (long long)blockIdx.x * (256LL * 4LL);
    v4f z = {0.0f, 0.0f, 0.0f, 0.0f};
#pragma unroll
    for (int k = 0; k < 4; ++k) {
        __builtin_nontemporal_store(z, out4 + base + (long long)k * 256 + tid);
    }
}

// ---------------------------------------------------------------------------
// Kernel 2: one wave (32 threads) per 16x16 tile.
// blockIdx.x: 0 = diagonal tile (j0 = i0), 1 = super-diagonal tile (j0 = i0+16)
// blockIdx.y: i-tile index (i0 = 16*y), blockIdx.z: batch b.
// Rank-1 outer product via V_WMMA_F32_16X16X4_F32 with K=1..3 zeroed:
//   A (16x4):  VGPR0 lanes 0-15 hold s[i0+lane] at K=0; all other slots 0.
//   B (4x16):  VGPR0 lanes 0-15 hold e[j0+lane] at K=0; all other slots 0.
//   C/D (16x16 f32): VGPR k -> M = k + 8*(lane>=16), N = lane%16.
// Band mask applied per element in registers, then coalesced dword stores.
// ---------------------------------------------------------------------------
__global__ void band_wmma_kernel(const float* __restrict__ s,
                                 const float* __restrict__ e,
                                 float* __restrict__ out) {
    const int lane = threadIdx.x;          // 0..31, one full wave
    const int i0   = blockIdx.y << 4;      // row-tile origin
    const int b    = blockIdx.z;
    const int j0   = i0 + (blockIdx.x << 4);
    if (j0 >= LEN) return;                 // wave-uniform: last super tile only

    const int m = lane & 15;

    // Unconditional loads (addresses valid for all lanes), then cndmask-select
    // so EXEC stays all-1s into the WMMA.
    float sv = s[b * LEN + i0 + m];
    float ev = e[b * LEN + j0 + m];
    float aval = (lane < 16) ? sv : 0.0f;  // zero K=2,3 half of A VGPR0
    float bval = (lane < 16) ? ev : 0.0f;  // zero K=2,3 half of B VGPR0

    v2f a = {aval, 0.0f};                  // VGPR1 = 0 -> K=1 (and K=3) zero
    v2f bm = {bval, 0.0f};                 // VGPR1 = 0 -> K=1 (and K=3) zero
    v8f c = {};

    // D = A x B + 0  (single multiply per element -> bit-exact vs scalar fp32)
    c = __builtin_amdgcn_wmma_f32_16x16x4_f32(
        /*neg_a=*/false, a, /*neg_b=*/false, bm,
        /*c_mod=*/(short)0, c, /*reuse_a=*/false, /*reuse_b=*/false);

    const int n     = lane & 15;
    const int mbase = (lane >= 16) ? 8 : 0;
#pragma unroll
    for (int k = 0; k < 8; ++k) {
        const int gi = i0 + mbase + k;
        const int gj = j0 + n;
        const bool band = (gj >= gi) && (gj - gi <= BAND);
        const float v = band ? c[k] : 0.0f;
        out[((long long)b * LEN + gi) * (long long)LEN + gj] = v;
    }
}

extern "C" void kernel_launch(void* const* d_in, const int* in_sizes, int n_in,
                              void* d_out, int out_size, void* d_ws, size_t ws_size,
                              hipStream_t stream) {
    const float* s = (const float*)d_in[0];
    const float* e = (const float*)d_in[1];
    float* out = (float*)d_out;

    // 1) Zero the full 512 MB output with NT 128-bit stores.
    const long long n4 = (long long)BATCH * LEN * LEN / 4;   // 33,554,432 float4
    const int zgrid = (int)(n4 / (256 * 4));                 // 32768 blocks
    zero_fill_kernel<<<zgrid, 256, 0, stream>>>((v4f*)out);

    // 2) Overwrite band tiles (same stream -> ordered after the fill).
    dim3 bgrid(2, LEN / 16, BATCH);                          // 2 x 128 x 32 waves
    band_wmma_kernel<<<bgrid, 32, 0, stream>>>(s, e, out);
}